// Net_14963666059852
// MI455X (gfx1250) — compile-verified
//
#include <hip/hip_runtime.h>

typedef __attribute__((ext_vector_type(2))) float v2f;
typedef __attribute__((ext_vector_type(4))) float v4f;
typedef __attribute__((ext_vector_type(8))) float v8f;

// ---------------------------------------------------------------------------
// Degree / normalization: deg = 1 + segment_sum(ones over dst); dis = rsqrt(deg)
// ---------------------------------------------------------------------------
__global__ void k_fill1(float* __restrict__ p, int n) {
  int i = blockIdx.x * blockDim.x + threadIdx.x;
  if (i < n) p[i] = 1.0f;
}

__global__ void k_count(const int* __restrict__ dst, float* __restrict__ deg, int e) {
  int i = blockIdx.x * blockDim.x + threadIdx.x;
  if (i < e) atomicAdd(&deg[dst[i]], 1.0f);
}

__global__ void k_rsqrt(float* __restrict__ p, int n) {
  int i = blockIdx.x * blockDim.x + threadIdx.x;
  if (i < n) p[i] = rsqrtf(p[i]);
}

// ---------------------------------------------------------------------------
// Pack W[K,N] into WMMA-fragment order WP[N,K]:
// per output column n, store all half-wave-0 K-pairs (k%4<2) in k order,
// then all half-wave-1 K-pairs (k%4>=2).  Each lane's B-fragment stream is
// then contiguous + 16B aligned -> one global_load_b128 feeds TWO k-steps.
// ---------------------------------------------------------------------------
__global__ void k_packW(const float* __restrict__ W, float* __restrict__ WP,
                        int K, int N) {
  int i = blockIdx.x * blockDim.x + threadIdx.x;
  if (i >= K * N) return;
  int k = i / N, n = i - k * N;
  int half = (k >> 1) & 1;                            // k%4 in {2,3} -> 1
  int pos = half * (K >> 1) + ((k >> 2) << 1) + (k & 1);
  WP[(size_t)n * K + pos] = W[i];
}

// ---------------------------------------------------------------------------
// WMMA fp32 GEMM:  H[M,N] = act(X)[M,K] @ W[K,N]   (act = optional ReLU)
// One wave -> 32 rows (2 m-tiles) x 128 cols (8 n-tiles).
// Per 8-wide K group: 8 b128 B loads + 4 ds_b64 A loads -> 32 WMMAs.
// Fragment layouts per CDNA5 ISA 7.12.2:
//   A 16x4 : lane%16 = M ; half-wave picks K pair (VGPR0=K0/K2, VGPR1=K1/K3)
//   B 4x16 : lane%16 = N ; half-wave picks K pair (fed from packed WP)
//   C/D    : lane%16 = N ; VGPR v -> M = v + 8*(lane/16)
// ---------------------------------------------------------------------------
template <int K, int N, bool RELU>
__global__ __launch_bounds__(256) void gemm_wmma_kernel(
    const float* __restrict__ X, const float* __restrict__ WP,  // WP packed [N,K]
    float* __restrict__ H, int M) {
  constexpr int NT = 8;        // n-tiles of 16 -> 128 cols per wave
  constexpr int MT = 2;        // m-tiles of 16 -> 32 rows per wave
  constexpr int NH = N / 128;  // number of 128-col column groups
  constexpr int KC = 32;       // K-chunk staged in LDS
  constexpr int LDR = KC + 4;  // padded row stride: 16B-aligned rows, bank-spread
  __shared__ float smem[8 * 32 * LDR];

  const int wave = threadIdx.x >> 5;
  const int lane = threadIdx.x & 31;
  const int wid = blockIdx.x * (blockDim.x >> 5) + wave;
  const int nh = wid % NH;
  const int mt = wid / NH;
  const int m0 = mt * 32;
  if (m0 + 32 > M) return;  // M is a multiple of 32 for this problem
  const int n0 = nh * 128;

  float* A = smem + wave * 32 * LDR;  // per-wave private LDS tile

  v8f acc[MT][NT];
#pragma unroll
  for (int t = 0; t < MT; ++t)
#pragma unroll
    for (int nt = 0; nt < NT; ++nt)
#pragma unroll
      for (int j = 0; j < 8; ++j) acc[t][nt][j] = 0.0f;

  const int am = lane & 15;         // A fragment: matrix row within tile
  const int ak = (lane >> 4) << 1;  // A fragment: K sub-offset (0 or 2)
  const int bn = lane & 15;         // B fragment: matrix col within tile

  // Single per-lane element offset into packed WP; everything else is
  // compile-time immediate or uniform scalar-base advance.
  const int vo = (n0 + bn) * K + (lane >> 4) * (K >> 1);

  // Staging geometry: 32 lanes x float4 cover 4 rows x KC floats per pass.
  const int sr = lane >> 3;          // 0..3
  const int sc = (lane & 7) << 2;    // 0,4,...,28
  const float* xp = X + (size_t)(m0 + sr) * K + sc;
  float* lp = A + sr * LDR + sc;     // 16B aligned (LDR, sc multiples of 4)

  for (int kc = 0; kc < K; kc += KC) {
    // ---- cooperative stage of the 32 x KC A-chunk (b128 in, b128 to LDS) ----
#pragma unroll
    for (int rr = 0; rr < 32; rr += 4) {
      float4 v = *(const float4*)(xp + (size_t)rr * K + kc);
      if (RELU) {
        v.x = fmaxf(v.x, 0.0f); v.y = fmaxf(v.y, 0.0f);
        v.z = fmaxf(v.z, 0.0f); v.w = fmaxf(v.w, 0.0f);
      }
      *(float4*)(lp + rr * LDR) = v;
    }
    // ---- math: 8-wide K groups; one b128 B load feeds 2 k-steps x 2 m-tiles --
    const float* wb = WP + (kc >> 1);  // uniform base advance (packed K/2 pitch)
#pragma unroll
    for (int k = 0; k < KC; k += 8) {
      v4f bq[NT];
#pragma unroll
      for (int nt = 0; nt < NT; ++nt)
        bq[nt] = *(const v4f*)(wb + vo + nt * 16 * K + (k >> 1));  // 16B aligned
#pragma unroll
      for (int s = 0; s < 2; ++s) {      // two 4-wide k-steps
        v2f a[MT];
#pragma unroll
        for (int t = 0; t < MT; ++t) {
          const float* ap = A + (t * 16 + am) * LDR + k + s * 4 + ak;
          a[t].x = ap[0];
          a[t].y = ap[1];
        }
#pragma unroll
        for (int nt = 0; nt < NT; ++nt) {
          v2f b;
          b.x = bq[nt][2 * s + 0];
          b.y = bq[nt][2 * s + 1];
#pragma unroll
          for (int t = 0; t < MT; ++t)
            acc[t][nt] = __builtin_amdgcn_wmma_f32_16x16x4_f32(
                false, a[t], false, b, (short)0, acc[t][nt], false, false);
        }
      }
    }
  }

  // ---- store: one base pointer per m-tile, all offsets immediate ----
  const int cn = lane & 15;
  const int rb = (lane >> 4) << 3;
#pragma unroll
  for (int t = 0; t < MT; ++t) {
    float* hp = H + (size_t)(m0 + t * 16 + rb) * N + n0 + cn;
#pragma unroll
    for (int nt = 0; nt < NT; ++nt)
#pragma unroll
      for (int v = 0; v < 8; ++v)
        hp[v * N + nt * 16] = acc[t][nt][v];
  }
}

// ---------------------------------------------------------------------------
// AGG init: AGG[m,f] = H[m,f] * dis[m]^2 + bias[f]  (self-loop + bias fused)
// ---------------------------------------------------------------------------
__global__ void k_agg_init(const float* __restrict__ H, const float* __restrict__ dis,
                           const float* __restrict__ bias, float* __restrict__ AGG,
                           int M, int F) {
  int i = blockIdx.x * blockDim.x + threadIdx.x;  // index in float4 units
  int total = (M * F) >> 2;
  if (i >= total) return;
  int f4 = i << 2;
  int m = f4 / F;
  int f = f4 - m * F;
  float d = dis[m];
  float dd = d * d;
  const float4 h = *(const float4*)(H + f4);
  const float4 b = *(const float4*)(bias + f);
  float4 r;
  r.x = fmaf(h.x, dd, b.x);
  r.y = fmaf(h.y, dd, b.y);
  r.z = fmaf(h.z, dd, b.z);
  r.w = fmaf(h.w, dd, b.w);
  *(float4*)(AGG + f4) = r;
}

// ---------------------------------------------------------------------------
// Edge scatter-add: AGG[dst,f] += H[src,f] * dis[src]*dis[dst]
// One edge per block (blockDim == F): coalesced reads + f32 atomics (L2).
// ---------------------------------------------------------------------------
__global__ void k_edge_agg(const float* __restrict__ H, const float* __restrict__ dis,
                           const int* __restrict__ src, const int* __restrict__ dst,
                           float* __restrict__ AGG, int F) {
  int e = blockIdx.x;
  int f = threadIdx.x;
  int s = src[e], d = dst[e];
  float nrm = dis[s] * dis[d];
  atomicAdd(&AGG[(size_t)d * F + f], H[(size_t)s * F + f] * nrm);
}

// ---------------------------------------------------------------------------
// Decode: one wave per eval edge; 128-ch dot via float4 + shfl_xor reduction.
// ---------------------------------------------------------------------------
__global__ void k_decode(const float* __restrict__ z, const int* __restrict__ pos,
                         const int* __restrict__ neg, float* __restrict__ out,
                         int npos, int nneg) {
  int w = (blockIdx.x * blockDim.x + threadIdx.x) >> 5;
  int lane = threadIdx.x & 31;
  int total = npos + nneg;
  if (w >= total) return;
  int a, b;
  if (w < npos) { a = pos[w];       b = pos[npos + w]; }
  else          { int j = w - npos; a = neg[j]; b = neg[nneg + j]; }
  const float4 za = *(const float4*)(z + (size_t)a * 128 + lane * 4);
  const float4 zb = *(const float4*)(z + (size_t)b * 128 + lane * 4);
  float s = za.x * zb.x + za.y * zb.y + za.z * zb.z + za.w * zb.w;
#pragma unroll
  for (int off = 16; off; off >>= 1) s += __shfl_xor(s, off, 32);
  if (lane == 0) out[w] = s;
}

// ---------------------------------------------------------------------------
extern "C" void kernel_launch(void* const* d_in, const int* in_sizes, int n_in,
                              void* d_out, int out_size, void* d_ws, size_t ws_size,
                              hipStream_t stream) {
  const float* x  = (const float*)d_in[0];
  const int*   ei = (const int*)d_in[1];
  const int*   pos = (const int*)d_in[2];
  const int*   neg = (const int*)d_in[3];
  const float* W1 = (const float*)d_in[4];
  const float* b1 = (const float*)d_in[5];
  const float* W2 = (const float*)d_in[6];
  const float* b2 = (const float*)d_in[7];
  const float* W3 = (const float*)d_in[8];
  const float* b3 = (const float*)d_in[9];
  float* out = (float*)d_out;

  const int M  = in_sizes[0] / 128;  // 100000 nodes
  const int E  = in_sizes[1] / 2;    // 1.6M edges
  const int NP = in_sizes[2] / 2;    // 200K pos eval edges
  const int NN = in_sizes[3] / 2;    // 200K neg eval edges

  const int* esrc = ei;
  const int* edst = ei + E;

  // Workspace: dis [M] | B0 [Mx256] | B1 [Mx256] | WP1 | WP2 | WP3
  char* ws = (char*)d_ws;
  float* dis = (float*)ws;
  size_t off = ((size_t)M * sizeof(float) + 255) & ~(size_t)255;
  size_t bsz = ((size_t)M * 256 * sizeof(float) + 255) & ~(size_t)255;
  float* B0  = (float*)(ws + off);
  float* B1  = (float*)(ws + off + bsz);
  float* WP1 = (float*)(ws + off + 2 * bsz);                      // 128*256
  float* WP2 = WP1 + 128 * 256;                                   // 256*256
  float* WP3 = WP2 + 256 * 256;                                   // 256*128

  dim3 blk(256);

  // Weight fragment-packing (tiny; L2-resident thereafter).
  k_packW<<<(128 * 256 + 255) / 256, blk, 0, stream>>>(W1, WP1, 128, 256);
  k_packW<<<(256 * 256 + 255) / 256, blk, 0, stream>>>(W2, WP2, 256, 256);
  k_packW<<<(256 * 128 + 255) / 256, blk, 0, stream>>>(W3, WP3, 256, 128);

  // degrees -> dis = rsqrt(1 + in-degree)
  k_fill1<<<(M + 255) / 256, blk, 0, stream>>>(dis, M);
  k_count<<<(E + 255) / 256, blk, 0, stream>>>(edst, dis, E);
  k_rsqrt<<<(M + 255) / 256, blk, 0, stream>>>(dis, M);

  const int mtiles32 = M / 32;               // 3125
  const int gb256 = (mtiles32 * 2 + 7) / 8;  // N=256: 2 column groups
  const int gb128 = (mtiles32 * 1 + 7) / 8;  // N=128: 1 column group

  // layer 1: H = X @ W1 ; AGG = norm-aggregate + bias
  gemm_wmma_kernel<128, 256, false><<<gb256, 256, 0, stream>>>(x, WP1, B0, M);
  k_agg_init<<<(M * 64 + 255) / 256, blk, 0, stream>>>(B0, dis, b1, B1, M, 256);
  k_edge_agg<<<E, 256, 0, stream>>>(B0, dis, esrc, edst, B1, 256);

  // layer 2 (ReLU fused on GEMM input)
  gemm_wmma_kernel<256, 256, true><<<gb256, 256, 0, stream>>>(B1, WP2, B0, M);
  k_agg_init<<<(M * 64 + 255) / 256, blk, 0, stream>>>(B0, dis, b2, B1, M, 256);
  k_edge_agg<<<E, 256, 0, stream>>>(B0, dis, esrc, edst, B1, 256);

  // layer 3 -> z in B1 [M x 128]
  gemm_wmma_kernel<256, 128, true><<<gb128, 256, 0, stream>>>(B1, WP3, B0, M);
  k_agg_init<<<(M * 32 + 255) / 256, blk, 0, stream>>>(B0, dis, b3, B1, M, 128);
  k_edge_agg<<<E, 128, 0, stream>>>(B0, dis, esrc, edst, B1, 128);

  // decode: per-edge dot products over pos ++ neg
  int total = NP + NN;
  k_decode<<<((size_t)total * 32 + 255) / 256, 256, 0, stream>>>(B1, pos, neg, out, NP, NN);
}